// Attention_26113401159791
// MI455X (gfx1250) — compile-verified
//
#include <hip/hip_runtime.h>
#include <hip/hip_bf16.h>
#include <stdint.h>

// Problem constants (match reference)
#define B_   4
#define C_   192
#define NH_  8
#define CH_  24        // C_/NH_
#define H_   128
#define W_   128
#define HW_  (H_ * W_) // 16384
#define C3_  (3 * C_)  // 576

typedef __attribute__((ext_vector_type(2))) float        v2f;
typedef __attribute__((ext_vector_type(8))) float        v8f;
typedef __attribute__((ext_vector_type(4))) unsigned int u32x4;
typedef __attribute__((ext_vector_type(8))) int          i32x8;
typedef __attribute__((ext_vector_type(4))) int          i32x4;

// ---------------------------------------------------------------------------
// Tensor Data Mover: 2D tile load (global -> LDS), fp32 elements.
// Packs D# group0/group1 per CDNA5 ISA 8.3/8.4; groups 2/3/(extra) zero.
// 6-arg builtin on this toolchain:
//   (uint32x4 g0, int32x8 g1, int32x4 g2, int32x4 g3, int32x8 extra, i32 cpol)
// ---------------------------------------------------------------------------
__device__ __forceinline__ void tdm_load_2d(unsigned lds_byte_addr,
                                            const void* gptr,
                                            unsigned tile_w, unsigned tile_h,
                                            unsigned long long row_stride_elems,
                                            unsigned tensor_w, unsigned tensor_h) {
  unsigned long long ga = (unsigned long long)gptr;
  u32x4 g0;
  g0[0] = 1u;                                              // count=1 (valid user D#)
  g0[1] = lds_byte_addr;                                   // lds_addr (bytes)
  g0[2] = (unsigned)(ga & 0xFFFFFFFFu);                    // global_addr[31:0]
  g0[3] = (unsigned)((ga >> 32) & 0x01FFFFFFu) | (2u << 30); // addr[56:32] | type=2
  i32x8 g1;
  g1[0] = (int)(2u << 16);                                 // data_size=2 -> 4 bytes
  g1[1] = (int)((tensor_w & 0xFFFFu) << 16);               // tensor_dim0[15:0]
  g1[2] = (int)(((tensor_w >> 16) & 0xFFFFu) | ((tensor_h & 0xFFFFu) << 16));
  g1[3] = (int)(((tensor_h >> 16) & 0xFFFFu) | ((tile_w & 0xFFFFu) << 16));
  g1[4] = (int)(tile_h & 0xFFFFu);                         // tile_dim1 (tile_dim2=0)
  g1[5] = (int)(row_stride_elems & 0xFFFFFFFFull);         // tensor_dim0_stride[31:0]
  g1[6] = (int)((row_stride_elems >> 32) & 0xFFFFull);     // stride[47:32] | dim1_stride lo = 0
  g1[7] = 0;
  i32x4 z4 = {0, 0, 0, 0};
  i32x8 z8 = {0, 0, 0, 0, 0, 0, 0, 0};
  __builtin_amdgcn_tensor_load_to_lds(g0, g1, z4, z4, z8, 0);
}

// Low 32 bits of the generic address of an LDS object = LDS byte address
// (aperture bits live in [63:32]).  The ptrtoint also marks the shared array
// as escaped so the compiler cannot fold its loads to undef.
__device__ __forceinline__ unsigned lds_addr_of(const void* p) {
  return (unsigned)(unsigned long long)(size_t)p;
}

// Compiler barrier: claims to write through p (TDM filled the LDS behind the
// compiler's back) so subsequent LDS reads must be materialized as ds_load.
__device__ __forceinline__ void lds_written(void* p) {
  asm volatile("" : "+v"(p) : : "memory");
}

// ---------------------------------------------------------------------------
// GEMM: out[b][m][n] = sum_k A[m][k] * Bm[b][k][n]  (+ add[b][m][n] if add)
// K = 192 fixed. Block = 128 thr = 4 waves; block tile 32(M) x 32(N);
// each wave computes one 16x16 tile with V_WMMA_F32_16X16X4_F32.
// A/B panels staged into LDS with the Tensor Data Mover.
// ---------------------------------------------------------------------------
__global__ __launch_bounds__(128)
void gemm_nn_tdm(const float* __restrict__ A,    // [Mtot][192]
                 const float* __restrict__ Bm,   // [B_][192][HW_]
                 const float* __restrict__ add,  // [B_][Mtot][HW_] or nullptr
                 float* __restrict__ out,        // [B_][Mtot][HW_]
                 int Mtot) {
  __shared__ float lds[12288];   // [0,6144): Btile [192][32]; [6144,12288): Atile [32][192]
  const int K = 192;
  const int n0 = blockIdx.x * 32;
  const int m0 = blockIdx.y * 32;
  const int b  = blockIdx.z;
  const int wv   = threadIdx.x >> 5;
  const int lane = threadIdx.x & 31;

  if (wv == 0) {
    const unsigned base = lds_addr_of(lds);
    // B panel: rows k=0..191, cols n0..n0+31
    tdm_load_2d(base, Bm + ((size_t)b * K) * HW_ + n0,
                32u, (unsigned)K, (unsigned long long)HW_,
                (unsigned)(HW_ - n0), (unsigned)K);
    // A panel: rows m0..m0+31, full K
    tdm_load_2d(base + 6144u * 4u, A + (size_t)m0 * K,
                (unsigned)K, 32u, (unsigned long long)K,
                (unsigned)K, (unsigned)(Mtot - m0));
    __builtin_amdgcn_s_wait_tensorcnt(0);
  }
  __syncthreads();
  lds_written((void*)lds);

  const float* Bt = lds;          // Bt[k*32 + n]
  const float* At = lds + 6144;   // At[m*192 + k]

  const int mt = (wv >> 1) * 16;
  const int nt = (wv & 1) * 16;
  const int lo = lane & 15;
  const int hi = lane >> 4;       // 0 or 1

  v8f acc = {};
  const float* Arow = At + (mt + lo) * K;
  const float* Bcol = Bt + (nt + lo);
#pragma unroll 4
  for (int k = 0; k < K; k += 4) {
    v2f av, bv;
    av = *(const v2f*)(Arow + k + 2 * hi);   // A[m, k+2*hi], A[m, k+2*hi+1]
    bv.x = Bcol[(k + 2 * hi + 0) * 32];      // B[k + 2*hi,     n]
    bv.y = Bcol[(k + 2 * hi + 1) * 32];      // B[k + 2*hi + 1, n]
    acc = __builtin_amdgcn_wmma_f32_16x16x4_f32(false, av, false, bv,
                                                (short)0, acc, false, false);
  }

  const int n = n0 + nt + lo;
#pragma unroll
  for (int i = 0; i < 8; ++i) {
    const int m = m0 + mt + i + 8 * hi;   // C/D layout: VGPR i -> M = i (+8 for hi lanes)
    const size_t oidx = ((size_t)b * Mtot + m) * HW_ + n;
    float v = acc[i];
    if (add) v += add[oidx];
    out[oidx] = v;
  }
}

// ---------------------------------------------------------------------------
// Depthwise 3x3, pad=1, groups=3C : qkv_dw = dwconv(qkv0)
// ---------------------------------------------------------------------------
__global__ __launch_bounds__(256)
void dwconv3x3(const float* __restrict__ src, const float* __restrict__ w,
               float* __restrict__ dst) {
  size_t idx = (size_t)blockIdx.x * 256 + threadIdx.x;
  if (idx >= (size_t)B_ * C3_ * HW_) return;
  const int x = (int)(idx % W_);
  const int h = (int)((idx / W_) % H_);
  const int c = (int)((idx / HW_) % C3_);
  const float* wp = w + c * 9;
  const float* sp = src + (idx - (size_t)(h * W_ + x)); // channel-plane base
  float s = 0.f;
#pragma unroll
  for (int dy = -1; dy <= 1; ++dy) {
    const int hh = h + dy;
    if (hh < 0 || hh >= H_) continue;
#pragma unroll
    for (int dx = -1; dx <= 1; ++dx) {
      const int xx = x + dx;
      if (xx < 0 || xx >= W_) continue;
      s += wp[(dy + 1) * 3 + (dx + 1)] * sp[hh * W_ + xx];
    }
  }
  dst[idx] = s;
}

// ---------------------------------------------------------------------------
// Conv branch fc: faithful raw-reshape gather.
// f_all[b,g,p,d] = qkv_flat_b[p*576 + g*24 + d]; f_conv[b, d*9+o, p] = fc(f_all)+b
// ---------------------------------------------------------------------------
__global__ __launch_bounds__(256)
void fc_kernel(const float* __restrict__ qkv, const float* __restrict__ fw,
               const float* __restrict__ fb, float* __restrict__ fconv) {
  const int t = blockIdx.x * 256 + threadIdx.x;
  if (t >= B_ * HW_ * CH_) return;
  const int d = t % CH_;
  const int p = (t / CH_) % HW_;
  const int b = t / (CH_ * HW_);
  const float* base = qkv + (size_t)b * C3_ * HW_ + (size_t)p * C3_ + d;
  float g[24];
#pragma unroll
  for (int gi = 0; gi < 24; ++gi) g[gi] = base[gi * CH_];
#pragma unroll
  for (int o = 0; o < 9; ++o) {
    float s = fb[o];
#pragma unroll
    for (int gi = 0; gi < 24; ++gi) s += fw[o * 24 + gi] * g[gi];
    fconv[((size_t)b * 216 + d * 9 + o) * HW_ + p] = s;
  }
}

// ---------------------------------------------------------------------------
// Grouped 3x3 conv: groups=24 (in/grp=9, out/grp=8), pad=1, + bias
// ---------------------------------------------------------------------------
__global__ __launch_bounds__(256)
void depconv3x3(const float* __restrict__ fconv, const float* __restrict__ dw,
                const float* __restrict__ db, float* __restrict__ outc) {
  size_t idx = (size_t)blockIdx.x * 256 + threadIdx.x;
  if (idx >= (size_t)B_ * C_ * HW_) return;
  const int x  = (int)(idx % W_);
  const int h  = (int)((idx / W_) % H_);
  const int oc = (int)((idx / HW_) % C_);
  const int b  = (int)(idx / ((size_t)HW_ * C_));
  const int grp = oc >> 3;                 // oc / (C_/CH_)
  float s = db[oc];
  const float* wbase = dw + oc * 81;       // dep_w[oc][j][ky][kx]
#pragma unroll
  for (int j = 0; j < 9; ++j) {
    const float* plane = fconv + ((size_t)b * 216 + grp * 9 + j) * HW_;
    const float* wj = wbase + j * 9;
#pragma unroll
    for (int dy = -1; dy <= 1; ++dy) {
      const int hh = h + dy;
      if (hh < 0 || hh >= H_) continue;
#pragma unroll
      for (int dx = -1; dx <= 1; ++dx) {
        const int xx = x + dx;
        if (xx < 0 || xx >= W_) continue;
        s += wj[(dy + 1) * 3 + (dx + 1)] * plane[hh * W_ + xx];
      }
    }
  }
  outc[idx] = s;
}

// ---------------------------------------------------------------------------
// Row L2 norms for q and k (rows 0..383 of qkv_dw per batch)
// ---------------------------------------------------------------------------
__global__ __launch_bounds__(256)
void rownorm_kernel(const float* __restrict__ qkv, float* __restrict__ norms) {
  __shared__ float red[256];
  const int row = blockIdx.x;               // [0, B_*2*C_)
  const int b = row / (2 * C_);
  const int c = row % (2 * C_);
  const float* p = qkv + ((size_t)b * C3_ + c) * HW_;
  float s = 0.f;
  for (int i = threadIdx.x; i < HW_; i += 256) { const float v = p[i]; s += v * v; }
  red[threadIdx.x] = s;
  __syncthreads();
  for (int st = 128; st > 0; st >>= 1) {
    if ((int)threadIdx.x < st) red[threadIdx.x] += red[threadIdx.x + st];
    __syncthreads();
  }
  if (threadIdx.x == 0) norms[row] = fmaxf(sqrtf(red[0]), 1e-12f);
}

// ---------------------------------------------------------------------------
// G[bh][c][d] = sum_n q[c,n]*k[d,n]  (unnormalized; norm folded in later)
// One block per (b,h); 4 waves cover the 24x24 output padded to 32x32.
// ---------------------------------------------------------------------------
__global__ __launch_bounds__(128)
void attn_qk(const float* __restrict__ qkv, float* __restrict__ G) {
  const int bh = blockIdx.x;
  const int b = bh >> 3, h = bh & 7;
  const int wv = threadIdx.x >> 5, lane = threadIdx.x & 31;
  const int mt = (wv >> 1) * 16, nt = (wv & 1) * 16;
  const int lo = lane & 15, hi = lane >> 4;
  const float* qb = qkv + ((size_t)b * C3_ + h * CH_) * HW_;
  const float* kb = qkv + ((size_t)b * C3_ + C_ + h * CH_) * HW_;
  const int mrow = mt + lo, ncol = nt + lo;
  const bool qv = mrow < CH_, kv = ncol < CH_;
  const float* qr = qb + (size_t)(qv ? mrow : 0) * HW_ + 2 * hi;
  const float* kr = kb + (size_t)(kv ? ncol : 0) * HW_ + 2 * hi;
  v8f acc = {};
  for (int n = 0; n < HW_; n += 4) {
    v2f av = {0.f, 0.f}, bv = {0.f, 0.f};
    if (qv) av = *(const v2f*)(qr + n);   // 8B-aligned global_load_b64
    if (kv) bv = *(const v2f*)(kr + n);
    acc = __builtin_amdgcn_wmma_f32_16x16x4_f32(false, av, false, bv,
                                                (short)0, acc, false, false);
  }
#pragma unroll
  for (int i = 0; i < 8; ++i) {
    const int ci = mt + i + 8 * hi, di = nt + lo;
    if (ci < CH_ && di < CH_) G[((size_t)bh * CH_ + ci) * CH_ + di] = acc[i];
  }
}

// ---------------------------------------------------------------------------
// attn[bh][c][d] = softmax_d( G * temp[h] / (||q_c|| * ||k_d||) )
// ---------------------------------------------------------------------------
__global__ __launch_bounds__(256)
void attn_softmax(const float* __restrict__ G, const float* __restrict__ norms,
                  const float* __restrict__ temp, float* __restrict__ attn) {
  const int t = blockIdx.x * 256 + threadIdx.x;
  if (t >= B_ * NH_ * CH_) return;
  const int bh = t / CH_, c = t % CH_;
  const int b = bh >> 3, h = bh & 7;
  const float nq = norms[b * 2 * C_ + h * CH_ + c];
  const float tm = temp[h];
  float row[24];
  float mx = -3.4e38f;
#pragma unroll
  for (int d = 0; d < CH_; ++d) {
    const float nk = norms[b * 2 * C_ + C_ + h * CH_ + d];
    row[d] = G[((size_t)bh * CH_ + c) * CH_ + d] * tm / (nq * nk);
    mx = fmaxf(mx, row[d]);
  }
  float sum = 0.f;
#pragma unroll
  for (int d = 0; d < CH_; ++d) { row[d] = __expf(row[d] - mx); sum += row[d]; }
  const float inv = 1.f / sum;
#pragma unroll
  for (int d = 0; d < CH_; ++d)
    attn[((size_t)bh * CH_ + c) * CH_ + d] = row[d] * inv;
}

// ---------------------------------------------------------------------------
// out_attn[b, h*24+c, n] = sum_d attn[bh][c][d] * v[bh][d][n]   (K=24)
// ---------------------------------------------------------------------------
__global__ __launch_bounds__(128)
void attn_v(const float* __restrict__ attn, const float* __restrict__ qkv,
            float* __restrict__ outa) {
  const int bh = blockIdx.y;
  const int b = bh >> 3, h = bh & 7;
  const int n0 = blockIdx.x * 32;
  const int wv = threadIdx.x >> 5, lane = threadIdx.x & 31;
  const int mt = (wv >> 1) * 16, nt = (wv & 1) * 16;
  const int lo = lane & 15, hi = lane >> 4;
  const float* vb = qkv + ((size_t)b * C3_ + 2 * C_ + h * CH_) * HW_;
  const float* ab = attn + (size_t)bh * CH_ * CH_;
  const int arow = mt + lo;
  const int n = n0 + nt + lo;
  v8f acc = {};
#pragma unroll
  for (int k = 0; k < CH_; k += 4) {
    const int kk = k + 2 * hi;
    v2f av = {0.f, 0.f}, bv;
    if (arow < CH_) av = *(const v2f*)(ab + arow * CH_ + kk);
    bv.x = vb[(size_t)kk * HW_ + n];
    bv.y = vb[(size_t)(kk + 1) * HW_ + n];
    acc = __builtin_amdgcn_wmma_f32_16x16x4_f32(false, av, false, bv,
                                                (short)0, acc, false, false);
  }
#pragma unroll
  for (int i = 0; i < 8; ++i) {
    const int ci = mt + i + 8 * hi;
    if (ci < CH_) outa[((size_t)b * C_ + h * CH_ + ci) * HW_ + n] = acc[i];
  }
}

// ---------------------------------------------------------------------------
// Host-side orchestration.
// Workspace layout (floats):
//   [0, szQ)          qkv0, later reused: fconv [0,szF) + outcv [szF, szF+szO)
//   [szQ, 2szQ)       qkv after depthwise (q|k|v rows)
//   [2szQ, 2szQ+szO)  out_attn
//   then G (18432), attn (18432), norms (1536)
// Peak = 2*37748736 + 12582912 + ~38.4K floats ≈ 352.5 MB of d_ws.
// ---------------------------------------------------------------------------
extern "C" void kernel_launch(void* const* d_in, const int* in_sizes, int n_in,
                              void* d_out, int out_size, void* d_ws, size_t ws_size,
                              hipStream_t stream) {
  const float* x      = (const float*)d_in[0];
  const float* qkv_w  = (const float*)d_in[1];
  const float* qkv_dw = (const float*)d_in[2];
  const float* proj_w = (const float*)d_in[3];
  const float* fc_w   = (const float*)d_in[4];
  const float* fc_b   = (const float*)d_in[5];
  const float* dep_w  = (const float*)d_in[6];
  const float* dep_b  = (const float*)d_in[7];
  const float* temp   = (const float*)d_in[8];
  float* out = (float*)d_out;
  float* ws  = (float*)d_ws;

  const size_t szQ = (size_t)B_ * C3_ * HW_;   // 37748736
  const size_t szO = (size_t)B_ * C_  * HW_;   // 12582912
  const size_t szF = (size_t)B_ * 216 * HW_;   // 14155776

  float* qkv0   = ws;
  float* qkvdw  = ws + szQ;
  float* outatt = ws + 2 * szQ;
  float* fconv  = ws;              // reuses qkv0 region (dead after dwconv)
  float* outcv  = ws + szF;        // still inside qkv0 region (szF+szO < szQ)
  float* Gbuf   = ws + 2 * szQ + szO;
  float* attn   = Gbuf + (size_t)B_ * NH_ * CH_ * CH_;
  float* norms  = attn + (size_t)B_ * NH_ * CH_ * CH_;

  // 1. qkv0 = qkv_w @ x            (WMMA f32 + TDM staging)
  gemm_nn_tdm<<<dim3(HW_ / 32, C3_ / 32, B_), 128, 0, stream>>>(
      qkv_w, x, nullptr, qkv0, C3_);
  // 2. depthwise 3x3
  dwconv3x3<<<(unsigned)((szQ + 255) / 256), 256, 0, stream>>>(qkv0, qkv_dw, qkvdw);
  // 3. fc (scrambled raw-reshape gather) -> f_conv
  fc_kernel<<<(B_ * HW_ * CH_ + 255) / 256, 256, 0, stream>>>(qkvdw, fc_w, fc_b, fconv);
  // 4. grouped 3x3 dep conv -> out_conv
  depconv3x3<<<(unsigned)((szO + 255) / 256), 256, 0, stream>>>(fconv, dep_w, dep_b, outcv);
  // 5. q/k row norms
  rownorm_kernel<<<B_ * 2 * C_, 256, 0, stream>>>(qkvdw, norms);
  // 6. G = q @ k^T                 (WMMA f32)
  attn_qk<<<B_ * NH_, 128, 0, stream>>>(qkvdw, Gbuf);
  // 7. normalize + temperature + softmax
  attn_softmax<<<(B_ * NH_ * CH_ + 255) / 256, 256, 0, stream>>>(Gbuf, norms, temp, attn);
  // 8. out_attn = attn @ v         (WMMA f32)
  attn_v<<<dim3(HW_ / 32, B_ * NH_), 128, 0, stream>>>(attn, qkvdw, outatt);
  // 9. out = proj_w @ out_attn + out_conv   (WMMA f32 + TDM staging)
  gemm_nn_tdm<<<dim3(HW_ / 32, C_ / 32, B_), 128, 0, stream>>>(
      proj_w, outatt, outcv, out, C_);
}